// PandaRodDifferentiableDynamicsChain_22823456211338
// MI455X (gfx1250) — compile-verified
//
#include <hip/hip_runtime.h>

typedef __attribute__((ext_vector_type(2))) float v2f;
typedef __attribute__((ext_vector_type(8))) float v8f;

#define JN 9
#define NEARZ 1e-6f

__device__ __forceinline__ void cross3(const float* a, const float* b, float* o) {
  o[0] = a[1]*b[2] - a[2]*b[1];
  o[1] = a[2]*b[0] - a[0]*b[2];
  o[2] = a[0]*b[1] - a[1]*b[0];
}

__device__ __forceinline__ void matvec3(const float* R, const float* x, float* o) {
#pragma unroll
  for (int r = 0; r < 3; ++r)
    o[r] = R[r*3+0]*x[0] + R[r*3+1]*x[1] + R[r*3+2]*x[2];
}

// o = R^T x
__device__ __forceinline__ void matvec3T(const float* R, const float* x, float* o) {
#pragma unroll
  for (int r = 0; r < 3; ++r)
    o[r] = R[0*3+r]*x[0] + R[1*3+r]*x[1] + R[2*3+r]*x[2];
}

// matrix_exp6(V) -> R (3x3), p (3); faithful to the JAX reference.
__device__ __forceinline__ void matexp6(const float* V, float* R, float* p) {
  float th = sqrtf(V[0]*V[0] + V[1]*V[1] + V[2]*V[2]);
  const bool near = th < NEARZ;
  float safe = near ? 1.0f : th;
  float inv  = 1.0f / safe;
  float u0 = V[0]*inv, u1 = V[1]*inv, u2 = V[2]*inv;
  float Wh[9] = {0.f, -u2, u1,  u2, 0.f, -u0,  -u1, u0, 0.f};
  float W2[9];
#pragma unroll
  for (int r = 0; r < 3; ++r)
#pragma unroll
    for (int c = 0; c < 3; ++c)
      W2[r*3+c] = Wh[r*3+0]*Wh[0*3+c] + Wh[r*3+1]*Wh[1*3+c] + Wh[r*3+2]*Wh[2*3+c];
  float st = sinf(th), ct = cosf(th);
  float omc = 1.0f - ct;
  float tms = th - st;
#pragma unroll
  for (int r = 0; r < 3; ++r) {
#pragma unroll
    for (int c = 0; c < 3; ++c) {
      float id  = (r == c) ? 1.0f : 0.0f;
      float val = id + st*Wh[r*3+c] + omc*W2[r*3+c];
      R[r*3+c]  = near ? id : val;
    }
  }
  float s0 = V[3]*inv, s1 = V[4]*inv, s2 = V[5]*inv;
#pragma unroll
  for (int r = 0; r < 3; ++r) {
    float g0 = ((r==0)?th:0.f) + omc*Wh[r*3+0] + tms*W2[r*3+0];
    float g1 = ((r==1)?th:0.f) + omc*Wh[r*3+1] + tms*W2[r*3+1];
    float g2 = ((r==2)?th:0.f) + omc*Wh[r*3+2] + tms*W2[r*3+2];
    float val = g0*s0 + g1*s1 + g2*s2;
    p[r] = near ? V[3+r] : val;
  }
}

// T_i = exp6(-A_i * q_i) @ trans_inv(Mlist[i]); returns R part and p part.
__device__ __forceinline__ void compute_T(int i, float qi,
                                          const float* __restrict__ Alist,
                                          const float* __restrict__ Mlist,
                                          float* Tr, float* Tp) {
  float V[6];
#pragma unroll
  for (int j = 0; j < 6; ++j) V[j] = -Alist[i*6+j] * qi;
  float Re[9], pe[3];
  matexp6(V, Re, pe);
  const float* M = Mlist + i*16;  // row-major 4x4
  float Rt[9];                    // = M_R^T
#pragma unroll
  for (int r = 0; r < 3; ++r)
#pragma unroll
    for (int c = 0; c < 3; ++c)
      Rt[r*3+c] = M[c*4+r];
  float pinv[3];
#pragma unroll
  for (int r = 0; r < 3; ++r)
    pinv[r] = -(Rt[r*3+0]*M[0*4+3] + Rt[r*3+1]*M[1*4+3] + Rt[r*3+2]*M[2*4+3]);
#pragma unroll
  for (int r = 0; r < 3; ++r) {
#pragma unroll
    for (int c = 0; c < 3; ++c)
      Tr[r*3+c] = Re[r*3+0]*Rt[0*3+c] + Re[r*3+1]*Rt[1*3+c] + Re[r*3+2]*Rt[2*3+c];
    Tp[r] = Re[r*3+0]*pinv[0] + Re[r*3+1]*pinv[1] + Re[r*3+2]*pinv[2] + pe[r];
  }
}

// Wave-cooperative y = G @ x per lane, G shared 6x6, via V_WMMA_F32_16X16X4_F32.
// A operands carry G in rows 0-5 (tile0: batch lanes 0-15) and duplicated in
// rows 8-13 (tile1: batch lanes 16-31), so after accumulating both tiles every
// lane reads its own 6 results directly from D VGPRs 0-5 (no output shuffles).
__device__ __forceinline__ void g_matvec_wmma(const float x[6], float y[6],
                                              int mrow, int half,
                                              v2f A00, v2f A01, v2f A10, v2f A11) {
  v8f acc = {0.f, 0.f, 0.f, 0.f, 0.f, 0.f, 0.f, 0.f};
#pragma unroll
  for (int t = 0; t < 2; ++t) {
    int s = mrow + 16*t;           // source batch lane for column (lane&15)
    float x0 = __shfl(x[0], s, 32);
    float x1 = __shfl(x[1], s, 32);
    float x2 = __shfl(x[2], s, 32);
    float x3 = __shfl(x[3], s, 32);
    float x4 = __shfl(x[4], s, 32);
    float x5 = __shfl(x[5], s, 32);
    // B 4x16 layout: lanes 0-15 hold K=0(V0),1(V1); lanes 16-31 hold K=2,3.
    v2f B0, B1;
    B0.x = half ? x2  : x0;  B0.y = half ? x3  : x1;   // K chunk 0 (cols 0-3)
    B1.x = half ? 0.f : x4;  B1.y = half ? 0.f : x5;   // K chunk 1 (cols 4-5)
    acc = __builtin_amdgcn_wmma_f32_16x16x4_f32(false, (t == 0) ? A00 : A10,
                                                false, B0, (short)0, acc,
                                                false, false);
    acc = __builtin_amdgcn_wmma_f32_16x16x4_f32(false, (t == 0) ? A01 : A11,
                                                false, B1, (short)0, acc,
                                                false, false);
  }
#pragma unroll
  for (int r = 0; r < 6; ++r) y[r] = acc[r];
}

__global__ __launch_bounds__(256) void inv_dyn_kernel(
    const float* __restrict__ q, const float* __restrict__ dq,
    const float* __restrict__ ddq, const float* __restrict__ g,
    const float* __restrict__ Ftip, const float* __restrict__ Alist,
    const float* __restrict__ Mlist, const float* __restrict__ Glist,
    float* __restrict__ tau, int B) {
  int gid = blockIdx.x * blockDim.x + threadIdx.x;
  int b = (gid < B) ? gid : (B - 1);   // clamp: keep EXEC all-ones for WMMA

  float lq[JN], ldq[JN], lddq[JN];
#pragma unroll
  for (int i = 0; i < JN; ++i) {
    lq[i]   = q[b*JN + i];
    ldq[i]  = dq[b*JN + i];
    lddq[i] = ddq[b*JN + i];
  }

  float v[6] = {0.f, 0.f, 0.f, 0.f, 0.f, 0.f};
  float a[6] = {0.f, 0.f, 0.f, -g[0], -g[1], -g[2]};
  float vs[JN][6], as_[JN][6];

  // ---------------- forward scan ----------------
#pragma unroll
  for (int i = 0; i < JN; ++i) {
    float Tr[9], Tp[3];
    compute_T(i, lq[i], Alist, Mlist, Tr, Tp);
    const float* A = Alist + i*6;
    float dqi = ldq[i], ddqi = lddq[i];

    // v_new = Ad(T) v + A dq   with Ad = [[R,0],[hat(p)R, R]]
    float Rvw[3], Rvv[3], pxw[3];
    matvec3(Tr, v, Rvw);
    matvec3(Tr, v + 3, Rvv);
    cross3(Tp, Rvw, pxw);
    float vn[6];
#pragma unroll
    for (int k = 0; k < 3; ++k) {
      vn[k]     = Rvw[k] + A[k]*dqi;
      vn[3 + k] = pxw[k] + Rvv[k] + A[3 + k]*dqi;
    }

    // a_new = Ad(T) a + ad(v_new) A dq + A ddq
    float Raw[3], Rav[3], pxa[3];
    matvec3(Tr, a, Raw);
    matvec3(Tr, a + 3, Rav);
    cross3(Tp, Raw, pxa);
    float t1[3], t2[3], t3[3];
    cross3(vn, A, t1);        // hat(w) Aw
    cross3(vn + 3, A, t2);    // hat(vv) Aw
    cross3(vn, A + 3, t3);    // hat(w) Av
    float an[6];
#pragma unroll
    for (int k = 0; k < 3; ++k) {
      an[k]     = Raw[k] + t1[k]*dqi + A[k]*ddqi;
      an[3 + k] = pxa[k] + Rav[k] + (t2[k] + t3[k])*dqi + A[3 + k]*ddqi;
    }
#pragma unroll
    for (int k = 0; k < 6; ++k) {
      vs[i][k] = vn[k]; as_[i][k] = an[k];
      v[k] = vn[k];     a[k] = an[k];
    }
  }

  // ---------------- backward scan ----------------
  const int lane = threadIdx.x & 31;
  const int mrow = lane & 15;
  const int half = lane >> 4;

  float F[6];
#pragma unroll
  for (int k = 0; k < 6; ++k) F[k] = Ftip[b*6 + k];

#pragma unroll
  for (int ii = JN - 1; ii >= 0; --ii) {
    // AdNext from T_{ii+1} (recomputed) or trans_inv(Mlist[JN]) for the last joint
    float Rn[9], Pn[3];
    if (ii == JN - 1) {
      const float* M = Mlist + JN*16;
#pragma unroll
      for (int r = 0; r < 3; ++r)
#pragma unroll
        for (int c = 0; c < 3; ++c)
          Rn[r*3+c] = M[c*4+r];
#pragma unroll
      for (int r = 0; r < 3; ++r)
        Pn[r] = -(Rn[r*3+0]*M[3] + Rn[r*3+1]*M[7] + Rn[r*3+2]*M[11]);
    } else {
      compute_T(ii + 1, lq[ii + 1], Alist, Mlist, Rn, Pn);
    }

    // AdN^T F = [ R^T (F_w - p x F_v) ; R^T F_v ]
    float cf[3];
    cross3(Pn, F + 3, cf);
    float tw[3] = {F[0] - cf[0], F[1] - cf[1], F[2] - cf[2]};
    float top[3], bot[3];
    matvec3T(Rn, tw, top);
    matvec3T(Rn, F + 3, bot);

    // Build WMMA A operands from shared G_i (padded 16x8, K split in two chunks;
    // A 16x4 layout: lanes 0-15 -> K=c0,c0+1 ; lanes 16-31 -> K=c0+2,c0+3)
    const float* G = Glist + ii*36;
    bool ok0 = (mrow < 6);
    bool ok1 = (mrow >= 8) && (mrow < 14);
    int  r0  = ok0 ? mrow : 0;
    int  r1  = ok1 ? (mrow - 8) : 0;
    int  c0  = 2*half;
    float ga = G[r0*6 + c0], gb = G[r0*6 + c0 + 1];
    float gc = G[r1*6 + c0], gd = G[r1*6 + c0 + 1];
    float ge = G[r0*6 + 4],  gf = G[r0*6 + 5];
    float gg = G[r1*6 + 4],  gh = G[r1*6 + 5];
    bool okh = (half == 0);  // chunk1 cols 4,5 live in lanes 0-15; 6,7 are padding
    v2f A00, A01, A10, A11;
    A00.x = ok0 ? ga : 0.f;           A00.y = ok0 ? gb : 0.f;
    A10.x = ok1 ? gc : 0.f;           A10.y = ok1 ? gd : 0.f;
    A01.x = (ok0 && okh) ? ge : 0.f;  A01.y = (ok0 && okh) ? gf : 0.f;
    A11.x = (ok1 && okh) ? gg : 0.f;  A11.y = (ok1 && okh) ? gh : 0.f;

    float Ga[6], Gv[6];
    g_matvec_wmma(as_[ii], Ga, mrow, half, A00, A01, A10, A11);
    g_matvec_wmma(vs[ii],  Gv, mrow, half, A00, A01, A10, A11);

    // F_new = AdN^T F + G a - ad(v)^T (G v)
    // ad(v)^T u = [ -(w x u_w + vv x u_v) ; -(w x u_v) ]
    float c1[3], c2[3], c3[3];
    cross3(vs[ii], Gv, c1);
    cross3(vs[ii] + 3, Gv + 3, c2);
    cross3(vs[ii], Gv + 3, c3);
    float Fn[6];
#pragma unroll
    for (int k = 0; k < 3; ++k) {
      Fn[k]     = top[k] + Ga[k]     + c1[k] + c2[k];
      Fn[3 + k] = bot[k] + Ga[3 + k] + c3[k];
    }

    const float* A = Alist + ii*6;
    float tv = Fn[0]*A[0] + Fn[1]*A[1] + Fn[2]*A[2] +
               Fn[3]*A[3] + Fn[4]*A[4] + Fn[5]*A[5];
    tau[b*JN + ii] = tv;
#pragma unroll
    for (int k = 0; k < 6; ++k) F[k] = Fn[k];
  }
}

extern "C" void kernel_launch(void* const* d_in, const int* in_sizes, int n_in,
                              void* d_out, int out_size, void* d_ws, size_t ws_size,
                              hipStream_t stream) {
  const float* q     = (const float*)d_in[0];
  const float* dq    = (const float*)d_in[1];
  const float* ddq   = (const float*)d_in[2];
  const float* g     = (const float*)d_in[3];
  const float* Ftip  = (const float*)d_in[4];
  const float* Alist = (const float*)d_in[5];
  const float* Mlist = (const float*)d_in[6];
  const float* Glist = (const float*)d_in[7];
  int B = in_sizes[0] / JN;
  if (B <= 0) return;
  dim3 block(256);
  dim3 grid((B + 255) / 256);
  inv_dyn_kernel<<<grid, block, 0, stream>>>(q, dq, ddq, g, Ftip,
                                             Alist, Mlist, Glist,
                                             (float*)d_out, B);
}